// BrillLindquistChristfSym_30511447670825
// MI455X (gfx1250) — compile-verified
//
#include <hip/hip_runtime.h>

// Brill-Lindquist Christoffel symbols on gfx1250 (MI455X).
// Bandwidth-bound: 12 B in / 108 B out per point (~125 MB total -> ~5.4 us
// at 23.3 TB/s). No GEMM structure -> WMMA inapplicable; optimize the
// CDNA5 data-movement paths instead:
//   - GLOBAL_LOAD_ASYNC_TO_LDS_B32 (ASYNCcnt) staging of the input tile
//   - LDS-staged output tile -> contiguous global_store_b128 th:NT
//   - global_prefetch_b8 for the next tile
//   - compile-time-P specialization to keep the VALU stream lean

typedef float v4f __attribute__((ext_vector_type(4)));

#define FD_EPS 1e-4f
#define NPTS 256                 // points per block = 8 wave32s
#define MAXP 8

#if defined(__AMDGCN__) && __has_builtin(__builtin_amdgcn_global_load_async_to_lds_b32)
#define HAVE_ASYNC_LDS 1
#else
#define HAVE_ASYNC_LDS 0
#endif

__device__ __forceinline__ void stage_b32(const float* g, float* l) {
#if HAVE_ASYNC_LDS
    // Builtin signature (from clang diagnostic): param 1 is AS1 int* (global
    // source), param 2 AS3 int* (LDS dest), then imm offset + imm cpol.
    // AS3 pointer value = low 32 bits of the flat LDS address (= LDS offset,
    // ISA 10.2 aperture mapping); AS1 value = global address.
    __builtin_amdgcn_global_load_async_to_lds_b32(
        (__attribute__((address_space(1))) int*)(uintptr_t)g,
        (__attribute__((address_space(3))) int*)(uintptr_t)l,
        /*offset=*/0, /*cpol=*/0);
#else
    *l = __builtin_nontemporal_load(g);
#endif
}

__device__ __forceinline__ void stage_wait() {
#if HAVE_ASYNC_LDS
#if __has_builtin(__builtin_amdgcn_s_wait_asynccnt)
    __builtin_amdgcn_s_wait_asynccnt(0);
#else
    asm volatile("s_wait_asynccnt 0x0" ::: "memory");
#endif
#endif
}

__device__ __forceinline__ float pow4f(float p) {
    float p2 = p * p;            // matches JAX integer_pow(psi,4): (x*x)*(x*x)
    return p2 * p2;
}

template <int NP, bool EXACT>
__device__ __forceinline__ float psi_eval(float px, float py, float pz,
                                          const float* cx, const float* cy,
                                          const float* cz, const float* ms,
                                          int P) {
    float s = 0.0f;
#pragma unroll
    for (int p = 0; p < NP; ++p) {
        if (EXACT || p < P) {
            float dx = px - cx[p];
            float dy = py - cy[p];
            float dz = pz - cz[p];
            // norm = sqrt(sum of squares); term = m/(2*r) (reference op order)
            float r = sqrtf(dx * dx + dy * dy + dz * dz);
            s += ms[p] / (2.0f * r);
        }
    }
    return 1.0f + s;
}

template <int NP, bool EXACT>
__global__ __launch_bounds__(NPTS) void
bl_christoffel_kernel(const float* __restrict__ xin,
                      const float* __restrict__ masses,
                      const float* __restrict__ centers,
                      float* __restrict__ out,
                      int B, int P) {
    __shared__ __align__(16) float xs[NPTS * 3];       // 3 KB
    __shared__ __align__(16) float stage[NPTS * 27];   // 27.6 KB

    const int tid = threadIdx.x;
    const int pbase = blockIdx.x * NPTS;
    int count = B - pbase;
    if (count > NPTS) count = NPTS;

    // ---- Phase 0: stage x tile into LDS (async-to-LDS when available) ----
    const float* xblk = xin + (size_t)pbase * 3;
    if (count == NPTS) {           // full block: no guards, 3 back-to-back copies
#pragma unroll
        for (int i = 0; i < 3; ++i) {
            int o = tid + i * NPTS;
            stage_b32(xblk + o, &xs[o]);
        }
    } else {
        const int nfl = count * 3;
#pragma unroll
        for (int i = 0; i < 3; ++i) {
            int o = tid + i * NPTS;
            if (o < nfl) stage_b32(xblk + o, &xs[o]);
        }
    }
    if (tid == 0 && pbase + NPTS < B) {
        __builtin_prefetch(xin + (size_t)(pbase + NPTS) * 3, 0, 0);  // global_prefetch_b8
    }

    // Uniform-address loads of masses/centers (tiny, cache-resident).
    float cx[NP], cy[NP], cz[NP], ms[NP];
#pragma unroll
    for (int p = 0; p < NP; ++p) {
        if (EXACT || p < P) {
            cx[p] = centers[3 * p + 0];
            cy[p] = centers[3 * p + 1];
            cz[p] = centers[3 * p + 2];
            ms[p] = masses[p];
        } else {
            cx[p] = 1.0f; cy[p] = 0.0f; cz[p] = 0.0f; ms[p] = 0.0f;
        }
    }

    stage_wait();        // drain ASYNCcnt before the barrier
    __syncthreads();

    // ---- Phase 1: one point per lane; 27 outputs into LDS stage ----
    if (tid < count) {
        float px = xs[3 * tid + 0];
        float py = xs[3 * tid + 1];
        float pz = xs[3 * tid + 2];

        float psi0 = psi_eval<NP, EXACT>(px, py, pz, cx, cy, cz, ms, P);
        float inv = 1.0f / pow4f(psi0);   // g^{kk} = psi^-4 (inv of diagonal)

        float F[3];
#pragma unroll
        for (int k = 0; k < 3; ++k) {
            float ax = px + (k == 0 ? FD_EPS : 0.0f);
            float ay = py + (k == 1 ? FD_EPS : 0.0f);
            float az = pz + (k == 2 ? FD_EPS : 0.0f);
            float bx = px - (k == 0 ? FD_EPS : 0.0f);
            float by = py - (k == 1 ? FD_EPS : 0.0f);
            float bz = pz - (k == 2 ? FD_EPS : 0.0f);
            float pp = psi_eval<NP, EXACT>(ax, ay, az, cx, cy, cz, ms, P);
            float pm = psi_eval<NP, EXACT>(bx, by, bz, cx, cy, cz, ms, P);
            float D = (pow4f(pp) - pow4f(pm)) / (2.0f * FD_EPS);
            F[k] = 0.5f * (inv * D);      // 0.5 * g^{kk} * d_k(psi^4)
        }
        float F0 = F[0], F1 = F[1], F2 = F[2];

        // C^k_ij = 0.5*(d_ik E_j + d_jk E_i - d_ij E_k); every entry reduces
        // EXACTLY (Sterbenz) to {+-F_t, 0}.  m = k*9 + i*3 + j.
        float* sp = &stage[tid * 27];     // stride 27 floats: coprime 64 banks
        sp[0]  = F0;   sp[1]  = F1;   sp[2]  = F2;    // k=0, i=0
        sp[3]  = F1;   sp[4]  = -F0;  sp[5]  = 0.0f;  // k=0, i=1
        sp[6]  = F2;   sp[7]  = 0.0f; sp[8]  = -F0;   // k=0, i=2
        sp[9]  = -F1;  sp[10] = F0;   sp[11] = 0.0f;  // k=1, i=0
        sp[12] = F0;   sp[13] = F1;   sp[14] = F2;    // k=1, i=1
        sp[15] = 0.0f; sp[16] = F2;   sp[17] = -F1;   // k=1, i=2
        sp[18] = -F2;  sp[19] = 0.0f; sp[20] = F0;    // k=2, i=0
        sp[21] = 0.0f; sp[22] = -F2;  sp[23] = F1;    // k=2, i=1
        sp[24] = F0;   sp[25] = F1;   sp[26] = F2;    // k=2, i=2
    }

    __syncthreads();

    // ---- Phase 2: coalesced b128 non-temporal streaming of the tile ----
    float* oblk = out + (size_t)pbase * 27;            // 16B-aligned
    const v4f* s4 = (const v4f*)stage;
    v4f* o4 = (v4f*)oblk;
    if (count == NPTS) {
        // 6912 floats = 1728 float4 = 6 full rounds + 192-thread tail
#pragma unroll
        for (int it = 0; it < 6; ++it) {
            int q = tid + it * NPTS;
            __builtin_nontemporal_store(s4[q], o4 + q); // global_store_b128 th:NT
        }
        int q = tid + 6 * NPTS;
        if (q < 1728) __builtin_nontemporal_store(s4[q], o4 + q);
    } else {
        const int nelem = count * 27;
        const int nf4 = nelem >> 2;
        for (int q = tid; q < nf4; q += NPTS) {
            __builtin_nontemporal_store(s4[q], o4 + q);
        }
        const int rem = nelem & 3;
        if (tid < rem) {
            int idx = nf4 * 4 + tid;
            __builtin_nontemporal_store(stage[idx], oblk + idx);
        }
    }
}

extern "C" void kernel_launch(void* const* d_in, const int* in_sizes, int n_in,
                              void* d_out, int out_size, void* d_ws, size_t ws_size,
                              hipStream_t stream) {
    const float* x = (const float*)d_in[0];        // [B,3]
    const float* masses = (const float*)d_in[1];   // [P]
    const float* centers = (const float*)d_in[2];  // [P,3]
    float* out = (float*)d_out;                    // [B,3,3,3]

    const int B = in_sizes[0] / 3;
    int P = in_sizes[1];
    if (P > MAXP) P = MAXP;

    const int grid = (B + NPTS - 1) / NPTS;
    switch (P) {
    case 1:
        bl_christoffel_kernel<1, true><<<grid, NPTS, 0, stream>>>(x, masses, centers, out, B, P);
        break;
    case 2:
        bl_christoffel_kernel<2, true><<<grid, NPTS, 0, stream>>>(x, masses, centers, out, B, P);
        break;
    case 3:
        bl_christoffel_kernel<3, true><<<grid, NPTS, 0, stream>>>(x, masses, centers, out, B, P);
        break;
    case 4:
        bl_christoffel_kernel<4, true><<<grid, NPTS, 0, stream>>>(x, masses, centers, out, B, P);
        break;
    default:
        bl_christoffel_kernel<MAXP, false><<<grid, NPTS, 0, stream>>>(x, masses, centers, out, B, P);
        break;
    }
}